// CrossAttention_33234456936515
// MI455X (gfx1250) — compile-verified
//
#include <hip/hip_runtime.h>
#include <stdint.h>

typedef float v2f __attribute__((ext_vector_type(2)));
typedef float v4f __attribute__((ext_vector_type(4)));
typedef float v8f __attribute__((ext_vector_type(8)));

// D = A(16x4) * B(4x16) + C, full fp32 matrix core (precision-safe for hard argmax attention)
#define WMMA_F32(a, b, c) \
  __builtin_amdgcn_wmma_f32_16x16x4_f32(false, (a), false, (b), (short)0, (c), false, false)

#define BATCH 4
#define NTOK 1024
#define QD 320
#define HEADS 8
#define DHEAD 64
#define INNER 512            // HEADS*DHEAD
#define SCALE_T 0.1875f      // 1.5 * 64^-0.5
#define NEGMAX -3.402823466e38f
#define KROW 68              // padded LDS row stride (floats): 16B-aligned, banks spread 4L

// ---------------------------------------------------------------------------
// Kernel 1: per-(batch, channel) mean / rstd over the token axis (biased var).
// grid = BATCH, block = QD (coalesced: consecutive threads -> consecutive c).
// ---------------------------------------------------------------------------
__global__ void ca_stats_kernel(const float* __restrict__ x,
                                float* __restrict__ mu, float* __restrict__ rstd) {
  const int b = blockIdx.x;
  const int c = threadIdx.x;                 // 0..319
  const float* xp = x + (size_t)b * NTOK * QD + c;
  float s = 0.f, s2 = 0.f;
  for (int n = 0; n < NTOK; ++n) {
    float v = xp[(size_t)n * QD];
    s += v; s2 += v * v;
  }
  const float m   = s  * (1.0f / NTOK);
  const float var = s2 * (1.0f / NTOK) - m * m;
  mu[b * QD + c]   = m;
  rstd[b * QD + c] = rsqrtf(var + 1e-5f);
}

// ---------------------------------------------------------------------------
// Kernel 2: qc[(b*8+h)][n][d] = 0.5*q_injected[h][n][d] + 0.5*(norm(x) @ W_q)
// GEMM M=4096, N=512, K=320 with V_WMMA_F32_16X16X4_F32; one wave per 16x16 tile.
// grid = 1024 blocks * 8 waves = 8192 tiles (256 M-tiles x 32 N-tiles).
// ---------------------------------------------------------------------------
__global__ void ca_qgemm_kernel(const float* __restrict__ x, const float* __restrict__ Wq,
                                const float* __restrict__ qinj, const float* __restrict__ mu,
                                const float* __restrict__ rstd, float* __restrict__ qc) {
  const int wave = threadIdx.x >> 5;
  const int lane = threadIdx.x & 31;
  const int tile = blockIdx.x * 8 + wave;
  const int mt = tile >> 5;                  // 0..255
  const int nt = tile & 31;                  // 0..31
  const int m0 = mt * 16, col0 = nt * 16;
  const int b  = m0 >> 10;                   // batch of this M tile
  const int hi = lane >> 4, lo = lane & 15;

  const float* arow  = x    + (size_t)(m0 + lo) * QD;  // A: lane = row (both halves)
  const float* murow = mu   + b * QD;
  const float* rsrow = rstd + b * QD;

  v8f c = {};
  for (int k = 0; k < QD; k += 4) {
    const int ka = k + 2 * hi;               // K pair owned by this lane half
    v2f xa = *(const v2f*)(arow + ka);
    v2f a;
    a.x = (xa.x - murow[ka])     * rsrow[ka];
    a.y = (xa.y - murow[ka + 1]) * rsrow[ka + 1];
    v2f bf;                                   // B: lane = column, strided K loads
    bf.x = Wq[(size_t)(ka)     * INNER + col0 + lo];
    bf.y = Wq[(size_t)(ka + 1) * INNER + col0 + lo];
    c = WMMA_F32(a, bf, c);
  }

  // scatter into [bh][n][d] layout, blended with injected Q
  const int h = col0 >> 6, dd0 = col0 & 63, ntok0 = m0 & (NTOK - 1);
  for (int r = 0; r < 8; ++r) {
    const int m = r + 8 * hi;                // C/D layout: vgpr r, half -> row
    const int ntok = ntok0 + m;
    const size_t qidx = ((size_t)h * NTOK + ntok) * DHEAD + dd0 + lo;
    const size_t oidx = ((size_t)(b * HEADS + h) * NTOK + ntok) * DHEAD + dd0 + lo;
    qc[oidx] = 0.5f * qinj[qidx] + 0.5f * c[r];
  }
}

// ---------------------------------------------------------------------------
// Kernel 3: fused masked QK^T + hard top-1 + V gather.
// Block = 8 waves, all on the same bh; K-tiles (16x64 fp32, 4KB) are staged in
// LDS with GLOBAL_LOAD_ASYNC_TO_LDS_B128 (ASYNCcnt), double-buffered so the
// async DMA of tile jt+1 overlaps the 16 fp32 WMMAs of tile jt. Q frags live
// in registers and are reused across all 64 key tiles. sim never materialized.
// grid = 256 blocks (32 bh x 8 i-tile groups) * 8 waves.
// ---------------------------------------------------------------------------
__global__ void ca_attn_kernel(const float* __restrict__ qc, const float* __restrict__ kinj,
                               const float* __restrict__ vinj, const float* __restrict__ mask,
                               float* __restrict__ attn_out) {
  __shared__ float sK[2][16 * KROW];         // 2 x 4.25KB ping-pong K tiles

  const int wave = threadIdx.x >> 5;
  const int lane = threadIdx.x & 31;
  const int bh = blockIdx.x >> 3;            // 0..31 (uniform across block)
  const int it = (blockIdx.x & 7) * 8 + wave;
  const int i0 = it * 16;
  const int h = bh & 7, b = bh >> 3;
  const int hi = lane >> 4, lo = lane & 15;

  const float* kbase = kinj + (size_t)h * NTOK * DHEAD;

  // Per-thread async-copy slot: 256 threads x 16B = one 4KB K tile.
  const int crow = threadIdx.x >> 4;         // 0..15 (key row within tile)
  const int cseg = threadIdx.x & 15;         // 0..15 (16B segment within row)
  const uint32_t ldsA = (uint32_t)(uintptr_t)(&sK[0][crow * KROW + cseg * 4]);
  const uint32_t ldsB = (uint32_t)(uintptr_t)(&sK[1][crow * KROW + cseg * 4]);
  const float* gsrc = kbase + (size_t)crow * DHEAD + cseg * 4;

  // Preload the 16 A-frags (Q tile rows, contiguous in K) — reused 64 times.
  v2f qa[16];
  const float* qrow = qc + ((size_t)bh * NTOK + i0 + lo) * DHEAD;
  for (int s = 0; s < 16; ++s)
    qa[s] = *(const v2f*)(qrow + 4 * s + 2 * hi);

  float bestv[8];
  int   besti[8];
  for (int r = 0; r < 8; ++r) { bestv[r] = NEGMAX; besti[r] = 0; }

  // Kick off tile 0 into buffer A.
  asm volatile("global_load_async_to_lds_b128 %0, %1, off"
               :: "v"(ldsA), "v"(gsrc) : "memory");

  for (int jt = 0; jt < 64; ++jt) {
    const int j0 = jt * 16;
    __syncthreads();                         // buf[(jt+1)&1] free (compute jt-1 done)
    if (jt < 63) {
      const uint32_t ldsn = ((jt + 1) & 1) ? ldsB : ldsA;
      const float* gn = gsrc + (size_t)(jt + 1) * 16 * DHEAD;
      asm volatile("global_load_async_to_lds_b128 %0, %1, off"
                   :: "v"(ldsn), "v"(gn) : "memory");
      asm volatile("s_wait_asynccnt 0x1" ::: "memory");  // my tile-jt segment done
      __builtin_prefetch(mask + (size_t)(i0 + 8 * hi) * NTOK + j0 + 16 + lo, 0, 0);
    } else {
      asm volatile("s_wait_asynccnt 0x0" ::: "memory");
    }
    __syncthreads();                         // whole tile jt resident in LDS

    // B-frag = k^T from LDS: lane = column j0+lo, contiguous K pair per half.
    const float* krow = &sK[jt & 1][lo * KROW];
    v8f c = {};
    for (int s = 0; s < 16; ++s) {
      v2f kb = *(const v2f*)(krow + 4 * s + 2 * hi);
      c = WMMA_F32(qa[s], kb, c);
    }
    // mask + running first-argmax (lane = column j0+lo, vgpr r + half = row)
    for (int r = 0; r < 8; ++r) {
      const int m = r + 8 * hi;
      const float msk = mask[(size_t)(i0 + m) * NTOK + j0 + lo];
      const float v = (msk < 0.5f) ? NEGMAX : c[r] * SCALE_T;
      if (v > bestv[r]) { bestv[r] = v; besti[r] = j0 + lo; }  // strict > keeps first index
    }
  }

  // Butterfly-reduce (max, first-index) across the 16 lanes of each half.
  for (int off = 1; off < 16; off <<= 1) {
    for (int r = 0; r < 8; ++r) {
      const float ov = __shfl_xor(bestv[r], off, 16);
      const int   oi = __shfl_xor(besti[r], off, 16);
      if (ov > bestv[r] || (ov == bestv[r] && oi < besti[r])) {
        bestv[r] = ov; besti[r] = oi;
      }
    }
  }

  // One-hot softmax => output row = v[argmax]. Gather + write in [b][n][h*d] layout.
  const float* vbase = vinj + (size_t)h * NTOK * DHEAD;
  for (int r = 0; r < 8; ++r) {
    const int m = r + 8 * hi;
    const int jstar = besti[r];
    v4f val = *(const v4f*)(vbase + (size_t)jstar * DHEAD + lo * 4);
    *(v4f*)(attn_out + ((size_t)(b * NTOK + i0 + m)) * INNER + h * DHEAD + lo * 4) = val;
  }
}

// ---------------------------------------------------------------------------
// Kernel 4: out = attn @ W_out + b_out.  M=4096, N=320, K=512.
// grid = 640 blocks * 8 waves = 5120 tiles (256 M-tiles x 20 N-tiles).
// ---------------------------------------------------------------------------
__global__ void ca_outgemm_kernel(const float* __restrict__ attn, const float* __restrict__ Wout,
                                  const float* __restrict__ bout, float* __restrict__ out) {
  const int wave = threadIdx.x >> 5;
  const int lane = threadIdx.x & 31;
  const int tile = blockIdx.x * 8 + wave;
  const int mt = tile / 20, nt = tile % 20;
  const int m0 = mt * 16, col0 = nt * 16;
  const int hi = lane >> 4, lo = lane & 15;

  const float* arow = attn + (size_t)(m0 + lo) * INNER;
  v8f c = {};
  for (int k = 0; k < INNER; k += 4) {
    const int ka = k + 2 * hi;
    v2f a = *(const v2f*)(arow + ka);
    v2f bf;
    bf.x = Wout[(size_t)(ka)     * QD + col0 + lo];
    bf.y = Wout[(size_t)(ka + 1) * QD + col0 + lo];
    c = WMMA_F32(a, bf, c);
  }
  const float bias = bout[col0 + lo];
  for (int r = 0; r < 8; ++r) {
    const int m = r + 8 * hi;
    out[(size_t)(m0 + m) * QD + col0 + lo] = c[r] + bias;
  }
}

// ---------------------------------------------------------------------------
extern "C" void kernel_launch(void* const* d_in, const int* in_sizes, int n_in,
                              void* d_out, int out_size, void* d_ws, size_t ws_size,
                              hipStream_t stream) {
  (void)in_sizes; (void)n_in; (void)out_size; (void)ws_size;
  const float* x    = (const float*)d_in[0];  // [4,1024,320]
  const float* qinj = (const float*)d_in[1];  // [8,1024,64]
  const float* kinj = (const float*)d_in[2];  // [8,1024,64]
  const float* vinj = (const float*)d_in[3];  // [8,1024,64]
  const float* imap = (const float*)d_in[4];  // [1,1024,1024]
  const float* Wq   = (const float*)d_in[5];  // [320,512]
  const float* Wout = (const float*)d_in[6];  // [512,320]
  const float* bout = (const float*)d_in[7];  // [320]
  float* out = (float*)d_out;                 // [4,1024,320]

  float* ws   = (float*)d_ws;
  float* mu   = ws;                                   // 1280 floats
  float* rstd = ws + 1280;                            // 1280 floats
  float* qc   = ws + 2560;                            // 32*1024*64 = 2M floats (8 MB)
  float* attn = qc + (size_t)32 * 1024 * 64;          // 4096*512  = 2M floats (8 MB)

  ca_stats_kernel  <<<BATCH, QD,  0, stream>>>(x, mu, rstd);
  ca_qgemm_kernel  <<<1024,  256, 0, stream>>>(x, Wq, qinj, mu, rstd, qc);
  ca_attn_kernel   <<<256,   256, 0, stream>>>(qc, kinj, vinj, imap, attn);
  ca_outgemm_kernel<<<640,   256, 0, stream>>>(attn, Wout, bout, out);
}